// EmbeddingShard_15539191677798
// MI455X (gfx1250) — compile-verified
//
#include <hip/hip_runtime.h>

#define N_VOCAB 50400
#define D_MODEL 4096
#define SHARDS  8
#define SEQ     2048
#define BATCH   4
#define VP      (N_VOCAB / SHARDS)   // 6300
#define TOKENS  (BATCH * SEQ)        // 8192

typedef float v4f __attribute__((ext_vector_type(4)));   // native vector: OK for NT builtins

// ---------------------------------------------------------------------------
// Main kernel: per token  out = kern[flat row x] + combo[sh_x] + pos[s]
// 256 threads/block (8 wave32s), one block per token.
//  - combo & pos rows (L2-resident) staged via CDNA5 async global->LDS DMA
//    (global_load_async_to_lds_b128, drained with s_wait_asynccnt).
//  - gather row (HBM, read-once) via non-temporal global_load_b128.
//  - output (write-once) via non-temporal stores.
// Each lane reads back only the LDS bytes it itself requested, so a
// wave-local s_wait_asynccnt suffices (no workgroup barrier needed).
// ---------------------------------------------------------------------------
__global__ void __launch_bounds__(256)
embed_main(const int*   __restrict__ xs,
           const float* __restrict__ kern,
           const float* __restrict__ combo,
           const float* __restrict__ pos,
           float*       __restrict__ out) {
  __shared__ float lds_combo[D_MODEL];
  __shared__ float lds_pos[D_MODEL];

  const int token = blockIdx.x;
  const int s     = token & (SEQ - 1);
  const int x     = xs[token];
  const int shx   = x / VP;

  const float* crow = combo + (size_t)shx * D_MODEL;
  const float* prow = pos   + (size_t)s   * D_MODEL;
  const float* krow = kern  + (size_t)x   * D_MODEL;
  float*       orow = out   + (size_t)token * D_MODEL;

  const int tid = threadIdx.x;

  // Async-DMA the two L2-resident rows into LDS (no VGPR landing zone).
#pragma unroll
  for (int i = 0; i < 4; ++i) {
    const int d = (tid + i * 256) * 4;                 // float index, 16B chunks
    unsigned lc = (unsigned)(size_t)(&lds_combo[d]);   // LDS byte address
    unsigned lp = (unsigned)(size_t)(&lds_pos[d]);
    asm volatile("global_load_async_to_lds_b128 %0, %1, off"
                 :: "v"(lc), "v"(crow + d) : "memory");
    asm volatile("global_load_async_to_lds_b128 %0, %1, off"
                 :: "v"(lp), "v"(prow + d) : "memory");
  }

  // Direct non-temporal loads of the gathered HBM row (overlaps the DMA).
  v4f kv[4];
#pragma unroll
  for (int i = 0; i < 4; ++i) {
    const int d = (tid + i * 256) * 4;
    kv[i] = __builtin_nontemporal_load((const v4f*)(krow + d));
  }

  // Drain the async-LDS counter, then combine from LDS and store.
  asm volatile("s_wait_asynccnt 0" ::: "memory");

#pragma unroll
  for (int i = 0; i < 4; ++i) {
    const int d = (tid + i * 256) * 4;
    const v4f c = *(const v4f*)(&lds_combo[d]);
    const v4f p = *(const v4f*)(&lds_pos[d]);
    v4f o = kv[i] + c + p;
    __builtin_nontemporal_store(o, (v4f*)(orow + d));
  }
}

// ---------------------------------------------------------------------------
// Kernel 1: combo[sh_x][d] = sum_sh bias[sh][d]
//                          + sum_{sh != sh_x} kernel[sh][ (sh<sh_x) ? VP-1 : 0 ][d]
// All 16 loads issued unconditionally up front -> single loadcnt wait,
// then a branch-free selected sum (shx is wave-uniform).
// ---------------------------------------------------------------------------
__global__ void __launch_bounds__(256)
build_combo(const float* __restrict__ kern,
            const float* __restrict__ bias,
            float* __restrict__ combo) {
  const int d   = blockIdx.x * blockDim.x + threadIdx.x;  // 0..4095
  const int shx = blockIdx.y;                             // 0..7

  float kv[SHARDS], bv[SHARDS];
#pragma unroll
  for (int sh = 0; sh < SHARDS; ++sh) {
    const int bnd = (sh < shx) ? (VP - 1) : 0;
    kv[sh] = kern[((size_t)sh * VP + bnd) * (size_t)D_MODEL + d];
    bv[sh] = bias[sh * D_MODEL + d];
  }
  float acc = 0.0f;
#pragma unroll
  for (int sh = 0; sh < SHARDS; ++sh) acc += bv[sh];
#pragma unroll
  for (int sh = 0; sh < SHARDS; ++sh)
    acc += (sh == shx) ? 0.0f : kv[sh];   // uniform select, no load under branch
  combo[(size_t)shx * D_MODEL + d] = acc;
}

// ---------------------------------------------------------------------------
// Fallback (no workspace): full 8-shard clipped gather per token.
// ---------------------------------------------------------------------------
__global__ void __launch_bounds__(256)
embed_full(const int*   __restrict__ xs,
           const float* __restrict__ kern,
           const float* __restrict__ bias,
           const float* __restrict__ pos,
           float*       __restrict__ out) {
  const int token = blockIdx.x;
  const int s     = token & (SEQ - 1);
  const int x     = xs[token];
  const int tid   = threadIdx.x;
  float*    orow  = out + (size_t)token * D_MODEL;

#pragma unroll
  for (int i = 0; i < 4; ++i) {
    const int d = (tid + i * 256) * 4;
    v4f acc = *(const v4f*)(pos + (size_t)s * D_MODEL + d);
#pragma unroll
    for (int sh = 0; sh < SHARDS; ++sh) {
      const v4f b = *(const v4f*)(bias + (size_t)sh * D_MODEL + d);
      int idx = x - sh * VP;
      idx = idx < 0 ? 0 : (idx > VP - 1 ? VP - 1 : idx);
      const v4f k =
          *(const v4f*)(kern + ((size_t)sh * VP + idx) * (size_t)D_MODEL + d);
      acc += b + k;
    }
    __builtin_nontemporal_store(acc, (v4f*)(orow + d));
  }
}

// ---------------------------------------------------------------------------
extern "C" void kernel_launch(void* const* d_in, const int* in_sizes, int n_in,
                              void* d_out, int out_size, void* d_ws, size_t ws_size,
                              hipStream_t stream) {
  const int*   xs   = (const int*)  d_in[0];   // [4,2048] int32
  const float* kern = (const float*)d_in[1];   // [8,6300,4096] f32
  const float* bias = (const float*)d_in[2];   // [8,4096] f32
  const float* pos  = (const float*)d_in[3];   // [8,2048,512] f32 (flat == [2048,4096])
  float*       out  = (float*)d_out;           // [4,2048,16,256] f32
  (void)in_sizes; (void)n_in; (void)out_size;

  const size_t combo_bytes = (size_t)SHARDS * D_MODEL * sizeof(float);
  if (ws_size >= combo_bytes) {
    float* combo = (float*)d_ws;
    build_combo<<<dim3(D_MODEL / 256, SHARDS), 256, 0, stream>>>(kern, bias, combo);
    embed_main<<<TOKENS, 256, 0, stream>>>(xs, kern, combo, pos, out);
  } else {
    embed_full<<<TOKENS, 256, 0, stream>>>(xs, kern, bias, pos, out);
  }
}